// MafiaGAT_68453188763878
// MI455X (gfx1250) — compile-verified
//
#include <hip/hip_runtime.h>
#include <math.h>

// ---------------- problem constants (from reference) ----------------
#define NNODES 50000
#define NEDGES 800000
#define ETOT   (NEDGES + NNODES)   // self-loops appended
#define IN_C   128
#define HIDW   32
#define OUT_C  8

typedef __attribute__((ext_vector_type(2))) float v2f;
typedef __attribute__((ext_vector_type(8))) float v8f;

// Order-preserving fp32 <-> uint mapping (for atomicMax-based segment max).
// enc is strictly monotone; enc(x) >= 1 for all finite x, so 0 == "-inf".
__device__ __forceinline__ unsigned enc_f32(float f) {
  unsigned u = __float_as_uint(f);
  return (u & 0x80000000u) ? ~u : (u | 0x80000000u);
}
__device__ __forceinline__ float dec_f32(unsigned u) {
  return __uint_as_float((u & 0x80000000u) ? (u & 0x7FFFFFFFu) : ~u);
}

__device__ __forceinline__ float elu_f(float v) {
  return v > 0.f ? v : (__expf(v) - 1.f);
}

// ---------------- dense GEMM via V_WMMA_F32_16X16X4_F32 ----------------
// Y[nrows x M] = X[nrows x K] @ W[K x M] + bias[M]
// One wave32 per 16x16 output tile; K stepped by 4 per WMMA.
// A frag: lane l -> row (l&15), VGPR v -> K = 2*(l>>4)+v  (ISA 7.12.2)
// B frag: lane l -> col (l&15), VGPR v -> K = 2*(l>>4)+v
// D frag: VGPR j -> row j + 8*(l>>4), col (l&15)
__global__ __launch_bounds__(32) void gemm_wmma_f32(
    const float* __restrict__ X, const float* __restrict__ W,
    const float* __restrict__ bias, float* __restrict__ Y,
    int K, int M)
{
  const int tilesN = M >> 4;
  const int tm = blockIdx.x / tilesN;
  const int tn = blockIdx.x % tilesN;
  const int lane = threadIdx.x & 31;
  const int half = lane >> 4;
  const int l16  = lane & 15;
  const int arow = tm * 16 + l16;
  const int bcol = tn * 16 + l16;
  const float* __restrict__ xrow = X + (size_t)arow * K;

  v8f acc = {};
  for (int k = 0; k < K; k += 4) {
    v2f a = *(const v2f*)(xrow + k + 2 * half);           // 8B-aligned
    v2f b;
    b.x = W[(size_t)(k + 2 * half)     * M + bcol];
    b.y = W[(size_t)(k + 2 * half + 1) * M + bcol];
    acc = __builtin_amdgcn_wmma_f32_16x16x4_f32(
        /*neg_a=*/false, a, /*neg_b=*/false, b,
        /*c_mod=*/(short)0, acc, /*reuse_a=*/false, /*reuse_b=*/false);
  }
  const float bv = bias[bcol];
#pragma unroll
  for (int j = 0; j < 8; ++j)
    Y[(size_t)(tm * 16 + j + 8 * half) * M + bcol] = acc[j] + bv;
}

// ---------------- edge_attr sum (for self-loop mean attr) ----------------
__global__ void sum_kernel(const float* __restrict__ a, float* __restrict__ out, int n) {
  int tid = blockIdx.x * blockDim.x + threadIdx.x;
  float s = 0.f;
  for (int i = tid; i < n; i += gridDim.x * blockDim.x) s += a[i];
#pragma unroll
  for (int off = 16; off > 0; off >>= 1) s += __shfl_down(s, off, 32);
  if ((threadIdx.x & 31) == 0) atomicAdd(out, s);
}

// ---------------- per-(edge,head) score + segment max ----------------
template <int HEADS>
__global__ __launch_bounds__(256) void edge_score_kernel(
    const int* __restrict__ srcI, const int* __restrict__ dstI,
    const float* __restrict__ eattr, const float* __restrict__ esum,
    const float* __restrict__ xl, const float* __restrict__ xr,
    const float* __restrict__ we, const float* __restrict__ att,
    float* __restrict__ score, unsigned* __restrict__ smax)
{
  const int C = HEADS * HIDW;
  int t = blockIdx.x * blockDim.x + threadIdx.x;
  if (t >= ETOT * HEADS) return;
  int e = t / HEADS, h = t % HEADS;
  int s, d; float ea;
  if (e < NEDGES) { s = srcI[e]; d = dstI[e]; ea = eattr[e]; }
  else            { s = d = e - NEDGES;       ea = esum[0] * (1.0f / NEDGES); }

  const float4* xls = (const float4*)(xl + (size_t)s * C + h * HIDW);
  const float4* xrd = (const float4*)(xr + (size_t)d * C + h * HIDW);
  const float4* weh = (const float4*)(we + h * HIDW);
  const float4* ath = (const float4*)(att + h * HIDW);
  float sc = 0.f;
#pragma unroll
  for (int q = 0; q < HIDW / 4; ++q) {
    float4 a = xls[q], b = xrd[q], w = weh[q], at = ath[q];
    float z;
    z = a.x + b.x + ea * w.x; z = z > 0.f ? z : 0.2f * z; sc += z * at.x;
    z = a.y + b.y + ea * w.y; z = z > 0.f ? z : 0.2f * z; sc += z * at.y;
    z = a.z + b.z + ea * w.z; z = z > 0.f ? z : 0.2f * z; sc += z * at.z;
    z = a.w + b.w + ea * w.w; z = z > 0.f ? z : 0.2f * z; sc += z * at.w;
  }
  score[t] = sc;
  atomicMax(&smax[(size_t)d * HEADS + h], enc_f32(sc));
}

// ---------------- exp(score - max) + segment sum ----------------
template <int HEADS>
__global__ __launch_bounds__(256) void edge_softmax_kernel(
    const int* __restrict__ dstI, const unsigned* __restrict__ smax,
    float* __restrict__ score, float* __restrict__ denom)
{
  int t = blockIdx.x * blockDim.x + threadIdx.x;
  if (t >= ETOT * HEADS) return;
  int e = t / HEADS, h = t % HEADS;
  int d = (e < NEDGES) ? dstI[e] : e - NEDGES;
  float mx = dec_f32(smax[(size_t)d * HEADS + h]);
  float ex = __expf(score[t] - mx);
  score[t] = ex;
  atomicAdd(&denom[(size_t)d * HEADS + h], ex);
}

// ---------------- weighted scatter-add: agg[dst] += alpha * xl[src] ----------------
template <int HEADS>
__global__ __launch_bounds__(256) void edge_aggregate_kernel(
    const int* __restrict__ srcI, const int* __restrict__ dstI,
    const float* __restrict__ score, const float* __restrict__ denom,
    const float* __restrict__ xl, float* __restrict__ agg)
{
  const int C = HEADS * HIDW;
  long long t = (long long)blockIdx.x * blockDim.x + threadIdx.x;
  if (t >= (long long)ETOT * C) return;
  int j = (int)(t % HIDW);
  int h = (int)((t / HIDW) % HEADS);
  int e = (int)(t / C);
  int s, d;
  if (e < NEDGES) { s = srcI[e]; d = dstI[e]; }
  else            { s = d = e - NEDGES; }
  float alpha = score[(size_t)e * HEADS + h] / denom[(size_t)d * HEADS + h];
  atomicAdd(&agg[(size_t)d * C + h * HIDW + j],
            xl[(size_t)s * C + h * HIDW + j] * alpha);
}

// ---------------- layer-1 finalize: h1 = elu(agg + bias), concat layout ----------------
__global__ __launch_bounds__(256) void finalize1_kernel(
    float* __restrict__ agg, const float* __restrict__ bias)
{
  int t = blockIdx.x * blockDim.x + threadIdx.x;
  if (t >= NNODES * 128) return;
  agg[t] = elu_f(agg[t] + bias[t & 127]);
}

// ---------------- layer-2 finalize: mean over 2 heads, +bias, elu ----------------
__global__ __launch_bounds__(256) void finalize2_kernel(
    const float* __restrict__ agg2, const float* __restrict__ bias,
    float* __restrict__ h2)
{
  int t = blockIdx.x * blockDim.x + threadIdx.x;
  if (t >= NNODES * HIDW) return;
  int n = t / HIDW, j = t % HIDW;
  float v = 0.5f * (agg2[(size_t)n * 64 + j] + agg2[(size_t)n * 64 + 32 + j]) + bias[j];
  h2[t] = elu_f(v);
}

// ---------------- classifier: out = h2 @ wc + bc (32 -> 8) ----------------
__global__ __launch_bounds__(256) void classifier_kernel(
    const float* __restrict__ h2, const float* __restrict__ wc,
    const float* __restrict__ bc, float* __restrict__ out)
{
  int t = blockIdx.x * blockDim.x + threadIdx.x;
  if (t >= NNODES * OUT_C) return;
  int n = t / OUT_C, o = t % OUT_C;
  const float* hr = h2 + (size_t)n * HIDW;
  float acc = bc[o];
#pragma unroll
  for (int j = 0; j < HIDW; ++j) acc += hr[j] * wc[j * OUT_C + o];
  out[t] = acc;
}

static inline int cdiv(long long a, int b) { return (int)((a + b - 1) / b); }

extern "C" void kernel_launch(void* const* d_in, const int* in_sizes, int n_in,
                              void* d_out, int out_size, void* d_ws, size_t ws_size,
                              hipStream_t stream)
{
  const float* x     = (const float*)d_in[0];
  const int*   eidx  = (const int*)d_in[1];
  const int*   srcI  = eidx;            // edge_index[0,:]
  const int*   dstI  = eidx + NEDGES;   // edge_index[1,:]
  const float* eattr = (const float*)d_in[2];
  const float* w1l = (const float*)d_in[3],  *b1l = (const float*)d_in[4];
  const float* w1r = (const float*)d_in[5],  *b1r = (const float*)d_in[6];
  const float* w1e = (const float*)d_in[7];
  const float* att1 = (const float*)d_in[8], *bias1 = (const float*)d_in[9];
  const float* w2l = (const float*)d_in[10], *b2l = (const float*)d_in[11];
  const float* w2r = (const float*)d_in[12], *b2r = (const float*)d_in[13];
  const float* w2e = (const float*)d_in[14];
  const float* att2 = (const float*)d_in[15], *bias2 = (const float*)d_in[16];
  const float* wc = (const float*)d_in[17],  *bc = (const float*)d_in[18];

  // ---- workspace layout (re-zeroed every call => deterministic) ----
  float* p = (float*)d_ws;
  float*    xl1  = p; p += (size_t)NNODES * 128;
  float*    xr1  = p; p += (size_t)NNODES * 128;
  float*    agg1 = p; p += (size_t)NNODES * 128;   // becomes h1 in place
  float*    score = p; p += (size_t)ETOT * 4;      // reused (layer2 needs ETOT*2)
  unsigned* smax = (unsigned*)p; p += (size_t)NNODES * 4;
  float*    denom = p; p += (size_t)NNODES * 4;
  float*    xl2  = p; p += (size_t)NNODES * 64;
  float*    xr2  = p; p += (size_t)NNODES * 64;
  float*    agg2 = p; p += (size_t)NNODES * 64;
  float*    h2   = p; p += (size_t)NNODES * 32;
  float*    esum = p; p += 1;
  size_t used = (size_t)((char*)p - (char*)d_ws);

  hipMemsetAsync(d_ws, 0, used, stream);

  // mean(edge_attr) for self-loop edge features
  sum_kernel<<<256, 256, 0, stream>>>(eattr, esum, NEDGES);

  // ---- layer 1: dense transforms (WMMA fp32) ----
  gemm_wmma_f32<<<(NNODES / 16) * (128 / 16), 32, 0, stream>>>(x, w1l, b1l, xl1, 128, 128);
  gemm_wmma_f32<<<(NNODES / 16) * (128 / 16), 32, 0, stream>>>(x, w1r, b1r, xr1, 128, 128);

  // ---- layer 1: edge phase (heads=4) ----
  edge_score_kernel<4><<<cdiv((long long)ETOT * 4, 256), 256, 0, stream>>>(
      srcI, dstI, eattr, esum, xl1, xr1, w1e, att1, score, smax);
  edge_softmax_kernel<4><<<cdiv((long long)ETOT * 4, 256), 256, 0, stream>>>(
      dstI, smax, score, denom);
  edge_aggregate_kernel<4><<<cdiv((long long)ETOT * 128, 256), 256, 0, stream>>>(
      srcI, dstI, score, denom, xl1, agg1);
  finalize1_kernel<<<cdiv((long long)NNODES * 128, 256), 256, 0, stream>>>(agg1, bias1);

  // ---- layer 2: dense transforms (WMMA fp32) ----
  gemm_wmma_f32<<<(NNODES / 16) * (64 / 16), 32, 0, stream>>>(agg1, w2l, b2l, xl2, 128, 64);
  gemm_wmma_f32<<<(NNODES / 16) * (64 / 16), 32, 0, stream>>>(agg1, w2r, b2r, xr2, 128, 64);

  // re-zero smax + denom (contiguous) for layer 2
  hipMemsetAsync(smax, 0, (size_t)NNODES * 4 * sizeof(unsigned) +
                          (size_t)NNODES * 4 * sizeof(float), stream);

  // ---- layer 2: edge phase (heads=2) ----
  edge_score_kernel<2><<<cdiv((long long)ETOT * 2, 256), 256, 0, stream>>>(
      srcI, dstI, eattr, esum, xl2, xr2, w2e, att2, score, smax);
  edge_softmax_kernel<2><<<cdiv((long long)ETOT * 2, 256), 256, 0, stream>>>(
      dstI, smax, score, denom);
  edge_aggregate_kernel<2><<<cdiv((long long)ETOT * 64, 256), 256, 0, stream>>>(
      srcI, dstI, score, denom, xl2, agg2);
  finalize2_kernel<<<cdiv((long long)NNODES * 32, 256), 256, 0, stream>>>(agg2, bias2, h2);

  // ---- classifier ----
  classifier_kernel<<<cdiv((long long)NNODES * 8, 256), 256, 0, stream>>>(
      h2, wc, bc, (float*)d_out);
}